// Yolov1Loss_1108101562907
// MI455X (gfx1250) — compile-verified
//
#include <hip/hip_runtime.h>
#include <hip/hip_bf16.h>

// YOLOv1 loss: memory-bound streaming reduction.
// 385 MB @ 23.3 TB/s -> ~16.5 us floor. NT loads (one-pass stream > 192MB L2),
// float2 loads (120B/cell -> 8B alignment), WMMA f32 16x16x4 for the
// cross-lane reduction, deterministic two-phase block reduction.

typedef __attribute__((ext_vector_type(2))) float v2f;
typedef __attribute__((ext_vector_type(8))) float v8f;

#define LAMBDA_COORD 5.0f
#define LAMBDA_NOOBJ 0.5f
#define YEPS 1e-6f

__device__ __forceinline__ float sqf(float x) { return x * x; }

__device__ __forceinline__ float iou4(float ax, float ay, float aw, float ah,
                                      float bx, float by, float bw, float bh) {
    float ax1 = ax - aw * 0.5f, ay1 = ay - ah * 0.5f;
    float ax2 = ax + aw * 0.5f, ay2 = ay + ah * 0.5f;
    float bx1 = bx - bw * 0.5f, by1 = by - bh * 0.5f;
    float bx2 = bx + bw * 0.5f, by2 = by + bh * 0.5f;
    float iw = fmaxf(fminf(ax2, bx2) - fmaxf(ax1, bx1), 0.0f);
    float ih = fmaxf(fminf(ay2, by2) - fmaxf(ay1, by1), 0.0f);
    float inter = iw * ih;
    float uni = (ax2 - ax1) * (ay2 - ay1) + (bx2 - bx1) * (by2 - by1) - inter;
    return inter / (uni + YEPS);
}

__global__ __launch_bounds__(256) void yolo_loss_partial(
    const float* __restrict__ pred, const float* __restrict__ targ,
    float* __restrict__ partials, int ncells) {
    int gid = blockIdx.x * blockDim.x + threadIdx.x;
    int stride = gridDim.x * blockDim.x;

    float acc = 0.0f;
    for (int cell = gid; cell < ncells; cell += stride) {
        const v2f* p2 = reinterpret_cast<const v2f*>(pred + (size_t)cell * 30);
        const v2f* t2 = reinterpret_cast<const v2f*>(targ + (size_t)cell * 30);
        float p[30], t[30];
#pragma unroll
        for (int j = 0; j < 15; ++j) {
            v2f v = __builtin_nontemporal_load(p2 + j);
            p[2 * j] = v.x; p[2 * j + 1] = v.y;
        }
#pragma unroll
        for (int j = 0; j < 15; ++j) {
            v2f v = __builtin_nontemporal_load(t2 + j);
            t[2 * j] = v.x; t[2 * j + 1] = v.y;
        }

        // responsible-box selection via IoU (comparison only)
        float i1 = iou4(p[1], p[2], p[3], p[4], t[1], t[2], t[3], t[4]);
        float i2 = iou4(p[6], p[7], p[8], p[9], t[6], t[7], t[8], t[9]);
        bool resp1 = i1 > i2;

        float bb1 = sqf(p[1] - t[1]) + sqf(p[2] - t[2]) +
                    sqf(sqrtf(p[3] + YEPS) - sqrtf(t[3] + YEPS)) +
                    sqf(sqrtf(p[4] + YEPS) - sqrtf(t[4] + YEPS));
        float bb2 = sqf(p[6] - t[6]) + sqf(p[7] - t[7]) +
                    sqf(sqrtf(p[8] + YEPS) - sqrtf(t[8] + YEPS)) +
                    sqf(sqrtf(p[9] + YEPS) - sqrtf(t[9] + YEPS));
        float bbox = LAMBDA_COORD * (resp1 ? bb1 : bb2);

        float d0 = sqf(p[0] - t[0]);
        float d5 = sqf(p[5] - t[5]);
        float conf_obj = resp1 ? (d0 + LAMBDA_NOOBJ * d5) : (d5 + LAMBDA_NOOBJ * d0);
        float conf_noobj = LAMBDA_NOOBJ * (d0 + d5);

        float cls = 0.0f;
#pragma unroll
        for (int e = 10; e < 30; ++e) cls += sqf(p[e] - t[e]);

        float per_cell = (t[0] == 1.0f) ? (bbox + conf_obj + cls) : conf_noobj;
        acc += per_cell;
    }

    // ---- Wave-level reduction on the matrix unit ----
    // D = A(16x4, all ones) x B(4x16) + C : every row of D holds the column
    // sums of B. We feed each lane's partial in B's first VGPR (second = 0);
    // regardless of the exact (k,n) lane mapping, summing one D-row over its
    // 16 columns yields the sum of all 32 lane partials. EXEC is all ones here.
    v2f a; a.x = 1.0f; a.y = 1.0f;
    v2f b; b.x = acc;  b.y = 0.0f;
    v8f c = {};
    c = __builtin_amdgcn_wmma_f32_16x16x4_f32(
        /*neg_a=*/false, a, /*neg_b=*/false, b,
        /*c_mod=*/(short)0, c, /*reuse_a=*/false, /*reuse_b=*/false);
    float s = c[0];
#pragma unroll
    for (int m = 1; m < 16; m <<= 1) s += __shfl_xor(s, m, 32);

    __shared__ float wsum[8];
    int lane = threadIdx.x & 31;
    int wave = threadIdx.x >> 5;
    if (lane == 0) wsum[wave] = s;
    __syncthreads();
    if (threadIdx.x == 0) {
        float bs = 0.0f;
#pragma unroll
        for (int w = 0; w < 8; ++w) bs += wsum[w];
        partials[blockIdx.x] = bs;
    }
}

__global__ __launch_bounds__(256) void yolo_loss_final(
    const float* __restrict__ partials, int n, float* __restrict__ out) {
    __shared__ float sm[256];
    float s = 0.0f;
    for (int i = threadIdx.x; i < n; i += 256) s += partials[i];
    sm[threadIdx.x] = s;
    __syncthreads();
    for (int off = 128; off > 0; off >>= 1) {
        if (threadIdx.x < off) sm[threadIdx.x] += sm[threadIdx.x + off];
        __syncthreads();
    }
    if (threadIdx.x == 0) out[0] = sm[0];
}

extern "C" void kernel_launch(void* const* d_in, const int* in_sizes, int n_in,
                              void* d_out, int out_size, void* d_ws, size_t ws_size,
                              hipStream_t stream) {
    const float* pred = (const float*)d_in[0];
    const float* targ = (const float*)d_in[1];
    float* out = (float*)d_out;
    float* partials = (float*)d_ws;

    int ncells = in_sizes[0] / 30;              // 32768*49 = 1,605,632
    int threads = 256;
    int blocks = (ncells + threads - 1) / threads;  // 6272
    int maxb = (int)(ws_size / sizeof(float));
    if (maxb > 0 && blocks > maxb) blocks = maxb;   // grid-stride covers rest
    if (blocks < 1) blocks = 1;

    yolo_loss_partial<<<blocks, threads, 0, stream>>>(pred, targ, partials, ncells);
    yolo_loss_final<<<1, 256, 0, stream>>>(partials, blocks, out);
}